// FastGPRLattice_72773925863687
// MI455X (gfx1250) — compile-verified
//
#include <hip/hip_runtime.h>
#include <math.h>

#ifndef M_PI
#define M_PI 3.14159265358979323846
#endif

// Problem constants (match the reference: N = 2^20, D = 8, sqrt(N) = 1024)
#define LATT_N   (1 << 20)
#define LOG2N    20
#define RSTRIDE  (LATT_N / 16)   // 65536: r-stride is invariant across Stockham stages (n*s == N)

typedef float v2f __attribute__((ext_vector_type(2)));
typedef float v8f __attribute__((ext_vector_type(8)));

// 16th roots of unity: ROOT16[k] = exp(-2*pi*i*k/16)  (DFT16 matrix entries)
__constant__ float2 ROOT16[16] = {
  { 1.00000000000000000f,  0.00000000000000000f},
  { 0.92387953251128674f, -0.38268343236508978f},
  { 0.70710678118654757f, -0.70710678118654746f},
  { 0.38268343236508984f, -0.92387953251128674f},
  { 0.00000000000000000f, -1.00000000000000000f},
  {-0.38268343236508973f, -0.92387953251128674f},
  {-0.70710678118654746f, -0.70710678118654757f},
  {-0.92387953251128674f, -0.38268343236508989f},
  {-1.00000000000000000f,  0.00000000000000000f},
  {-0.92387953251128685f,  0.38268343236508967f},
  {-0.70710678118654768f,  0.70710678118654746f},
  {-0.38268343236508995f,  0.92387953251128663f},
  { 0.00000000000000000f,  1.00000000000000000f},
  { 0.38268343236508956f,  0.92387953251128685f},
  { 0.70710678118654735f,  0.70710678118654768f},
  { 0.92387953251128663f,  0.38268343236509000f},
};

__device__ __forceinline__ unsigned bitrev20(unsigned j) {
  return __brev(j) >> (32 - LOG2N);
}

// ---------------------------------------------------------------------------
// Prep: k1br[j] = gs * prod_d(1 + ls[d]*k1full[BR(j),d]) (+noise at 0),
//       ybr[j]  = y[BR(j)].  Fuses the bit-reversal permutation so the FFT
//       can run natural-order Stockham.  f64 in -> f32 complex out.
// ---------------------------------------------------------------------------
__global__ void gpr_prep_kernel(const double* __restrict__ rgs,
                                const double* __restrict__ rls,
                                const double* __restrict__ rnoise,
                                const double* __restrict__ k1full,
                                const double* __restrict__ y,
                                float2* __restrict__ W0) {
  const int j = blockIdx.x * blockDim.x + threadIdx.x;
  const unsigned i = bitrev20((unsigned)j);
  const double gs    = exp(rgs[0]);
  const double noise = exp(rnoise[0]);
  const double* row = k1full + (size_t)i * 8;
  double prod = 1.0;
#pragma unroll
  for (int d = 0; d < 8; ++d)
    prod *= 1.0 + exp(rls[d]) * row[d];
  const double k1 = gs * prod + (j == 0 ? noise : 0.0);  // BR(0)==0
  W0[j]          = make_float2((float)k1, 0.0f);
  W0[LATT_N + j] = make_float2((float)y[i], 0.0f);
}

// ---------------------------------------------------------------------------
// One radix-16 Stockham stage on the matrix core.
// Stage t: s = 16^t (passed as l2s = 4t), nsub = N >> l2s, m = nsub/16.
// For column L = (p,q):
//   out[q + s*(16p+u)] = w_nsub^{p*u} * sum_r in[q + s*p + r*(N/16)] * w16^{u*r}
// Each wave handles one 16(row r) x 16(column) complex tile:
//   D = A*B as 4 real 16x16x16 matmuls = 16 x v_wmma_f32_16x16x4_f32.
// A-operand layout per ISA (16x4 f32): lanes 0-15 / 16-31 hold M=0..15,
//   VGPR0 = K{0|2}, VGPR1 = K{1|3}; B mirrors (rows split by lane-half).
// DFT16 entries come from the ROOT16 constant table (single cache line);
// only the per-column Stockham twiddles use the transcendental pipe (which
// co-executes with VALU/WMMA).  EXEC is all-ones: full blocks, no divergence.
// ---------------------------------------------------------------------------
__global__ void fft16_stage_kernel(const float2* __restrict__ src,
                                   float2* __restrict__ dst,
                                   int l2s, float inv_nsub) {
  const int lane   = threadIdx.x & 31;
  const int wid    = threadIdx.x >> 5;
  const int half   = lane >> 4;         // lane half (0/1)
  const int lane16 = lane & 15;
  const int tile   = blockIdx.x * (blockDim.x >> 5) + wid;
  const long boff  = (long)blockIdx.y * LATT_N;     // batch offset

  const int s = 1 << l2s;

  // column owned by this lane (both halves share it for B/C/D)
  const int  L     = tile * 16 + lane16;            // in [0, N/16)
  const int  p     = L >> l2s;
  const int  q     = L & (s - 1);
  const long cbase = boff + q + (long)s * p;

  // ---- B operand: 16x16 complex data tile (rows r across K-chunks) ----
  v2f br[4], bi[4];
#pragma unroll
  for (int kk = 0; kk < 4; ++kk) {
    const int r0 = 4 * kk + 2 * half;               // rows r0, r0+1 for this half
    const float2 e0 = src[cbase + (long)r0       * RSTRIDE];
    const float2 e1 = src[cbase + (long)(r0 + 1) * RSTRIDE];
    br[kk].x = e0.x; br[kk].y = e1.x;
    bi[kk].x = e0.y; bi[kk].y = e1.y;
  }

  // ---- A operand: DFT16, A[u][k] = ROOT16[(u*k) mod 16], u = lane16 ----
  // f32 WMMA has no A/B negation (NEG = CNeg only), so materialize -Im(A).
  v2f ar[4], ai[4], aim[4];
  const int u = lane16;
#pragma unroll
  for (int kk = 0; kk < 4; ++kk) {
    const int k0 = 4 * kk + 2 * half;
    const float2 w0 = ROOT16[(u * k0)       & 15];
    const float2 w1 = ROOT16[(u * (k0 + 1)) & 15];
    ar[kk].x  =  w0.x; ar[kk].y  =  w1.x;
    ai[kk].x  =  w0.y; ai[kk].y  =  w1.y;
    aim[kk].x = -w0.y; aim[kk].y = -w1.y;
  }

  // ---- complex 16x16x16 matmul: Dr = ArBr - AiBi, Di = ArBi + AiBr ----
  v8f cr = {};
  v8f ci = {};
#pragma unroll
  for (int kk = 0; kk < 4; ++kk) {
    cr = __builtin_amdgcn_wmma_f32_16x16x4_f32(false, ar[kk],  false, br[kk], (short)0, cr, false, false);
    cr = __builtin_amdgcn_wmma_f32_16x16x4_f32(false, aim[kk], false, bi[kk], (short)0, cr, false, false);
    ci = __builtin_amdgcn_wmma_f32_16x16x4_f32(false, ar[kk],  false, bi[kk], (short)0, ci, false, false);
    ci = __builtin_amdgcn_wmma_f32_16x16x4_f32(false, ai[kk],  false, br[kk], (short)0, ci, false, false);
  }

  // ---- per-output twiddle w_nsub^{p*u} on the trans pipe, Stockham scatter ----
  // C/D layout: VGPR v holds row M = v + 8*half, col N = lane16.
#pragma unroll
  for (int v = 0; v < 8; ++v) {
    const int   uo   = v + 8 * half;
    const float frac = (float)(p * uo) * inv_nsub;  // p*uo < nsub <= 2^20: exact in fp32
    float sn, cs;
    __sincosf(-2.0f * (float)M_PI * frac, &sn, &cs);
    const float dr = cr[v] * cs - ci[v] * sn;
    const float di = cr[v] * sn + ci[v] * cs;
    dst[boff + q + (long)(16 * p + uo) * s] = make_float2(dr, di);
  }
}

// ---------------------------------------------------------------------------
// Mid: lam = F(k1br), ytilde = F(ybr)/sqrt(N).
//   w = conj(ytilde/lam)  (input to FFT #2, since ifft(z)=conj(fft(conj z))/N)
//   accumulate sum Re(ytilde^2 * lam) and sum log|lam| (f64 reduction).
// ---------------------------------------------------------------------------
__global__ void gpr_mid_kernel(const float2* __restrict__ W1,
                               float2* __restrict__ W0,
                               double* __restrict__ acc) {
  const int j = blockIdx.x * blockDim.x + threadIdx.x;
  const float2 lam = W1[j];
  const float2 ytr = W1[LATT_N + j];
  const float invsq = 1.0f / 1024.0f;               // 1/sqrt(N)
  const float a = ytr.x * invsq, b = ytr.y * invsq; // ytilde
  const float c = lam.x, d = lam.y;
  const float mag2 = c * c + d * d;
  const float ur = (a * c + b * d) / mag2;          // ytilde/lam
  const float ui = (b * c - a * d) / mag2;
  W0[j] = make_float2(ur, -ui);                     // conj

  double quad = (double)((a * a - b * b) * c - 2.0f * a * b * d); // Re(yt^2*lam)
  double ldet = 0.5 * (double)logf(mag2);                          // log|lam|

  __shared__ double sq[256], sl[256];
  sq[threadIdx.x] = quad; sl[threadIdx.x] = ldet;
  __syncthreads();
  for (int off = 128; off > 0; off >>= 1) {
    if (threadIdx.x < off) {
      sq[threadIdx.x] += sq[threadIdx.x + off];
      sl[threadIdx.x] += sl[threadIdx.x + off];
    }
    __syncthreads();
  }
  if (threadIdx.x == 0) {
    atomicAdd(&acc[0], sq[0]);
    atomicAdd(&acc[1], sl[0]);
  }
}

__global__ void gpr_zero_kernel(double* __restrict__ acc) {
  if (threadIdx.x < 2) acc[threadIdx.x] = 0.0;
}

// ---------------------------------------------------------------------------
// Final: coeffs[j] = conj(F[BR(j)]) / sqrt(N);  out[0] = mll.
// ---------------------------------------------------------------------------
__global__ void gpr_final_kernel(const float2* __restrict__ F,
                                 const double* __restrict__ acc,
                                 float* __restrict__ out, int out_size) {
  const int j = blockIdx.x * blockDim.x + threadIdx.x;
  if (j == 0 && out_size > 0) {
    const double mll = acc[0] - acc[1] - (double)LATT_N * log(2.0 * M_PI);
    out[0] = (float)mll;
  }
  const unsigned k = bitrev20((unsigned)j);
  const float2 f = F[k];
  const float invsq = 1.0f / 1024.0f;
  const int i1 = 1 + 2 * j, i2 = 2 + 2 * j;
  if (i2 < out_size) {
    out[i1] =  f.x * invsq;
    out[i2] = -f.y * invsq;
  }
}

// ---------------------------------------------------------------------------
// Host-side launch.  Workspace layout (needs >= 32 MB + 16 B):
//   W0: 2N complex64 (16 MB)  |  W1: 2N complex64 (16 MB)  |  acc: 2 doubles
// ---------------------------------------------------------------------------
extern "C" void kernel_launch(void* const* d_in, const int* in_sizes, int n_in,
                              void* d_out, int out_size, void* d_ws, size_t ws_size,
                              hipStream_t stream) {
  (void)in_sizes; (void)n_in; (void)ws_size;
  const double* rgs    = (const double*)d_in[0];
  const double* rls    = (const double*)d_in[1];
  const double* rnoise = (const double*)d_in[2];
  const double* k1full = (const double*)d_in[3];
  const double* y      = (const double*)d_in[4];
  float* out = (float*)d_out;

  char* ws = (char*)d_ws;
  float2* W0  = (float2*)ws;
  float2* W1  = (float2*)(ws + (size_t)2 * LATT_N * sizeof(float2));
  double* acc = (double*)(ws + (size_t)4 * LATT_N * sizeof(float2));

  const int threads = 256;                    // 8 waves/block (wave32)
  const int nblk    = LATT_N / threads;       // 4096
  const int fftblk  = (LATT_N / 256) / 8;     // 512 blocks: 1 tile (16x16) per wave

  gpr_zero_kernel<<<1, 64, 0, stream>>>(acc);
  gpr_prep_kernel<<<nblk, threads, 0, stream>>>(rgs, rls, rnoise, k1full, y, W0);

  // FFT #1: batched-2 (k1br || ybr), 5 radix-16 Stockham stages, W0<->W1.
  for (int t = 0; t < 5; ++t) {
    const float2* s = (t & 1) ? W1 : W0;
    float2*       d = (t & 1) ? W0 : W1;
    const float inv_nsub = 1.0f / (float)(LATT_N >> (4 * t));
    fft16_stage_kernel<<<dim3(fftblk, 2), threads, 0, stream>>>(s, d, 4 * t, inv_nsub);
  }
  // W1 now holds lam=[0,N) and ytilde_raw=[N,2N).
  gpr_mid_kernel<<<nblk, threads, 0, stream>>>(W1, W0, acc);

  // FFT #2: single array w = conj(ytilde/lam), ends in W1.
  for (int t = 0; t < 5; ++t) {
    const float2* s = (t & 1) ? W1 : W0;
    float2*       d = (t & 1) ? W0 : W1;
    const float inv_nsub = 1.0f / (float)(LATT_N >> (4 * t));
    fft16_stage_kernel<<<dim3(fftblk, 1), threads, 0, stream>>>(s, d, 4 * t, inv_nsub);
  }
  gpr_final_kernel<<<nblk, threads, 0, stream>>>(W1, acc, out, out_size);
}